// K1CompleteSystem_55576876810843
// MI455X (gfx1250) — compile-verified
//
#include <hip/hip_runtime.h>

typedef float v2f __attribute__((ext_vector_type(2)));
typedef float v8f __attribute__((ext_vector_type(8)));

#define A_AG 21
#define DD   128
#define HH   256
#define VV   32000
#define NN   4096          // 8 * 512 tokens
#define EPSF 1e-5f

__device__ __forceinline__ v8f wmma_f32(v2f a, v2f b, v8f c) {
    // D = A(16x4 f32) * B(4x16 f32) + C(16x16 f32)
    return __builtin_amdgcn_wmma_f32_16x16x4_f32(
        /*neg_a=*/false, a, /*neg_b=*/false, b,
        /*c_mod=*/(short)0, c, /*reuse_a=*/false, /*reuse_b=*/false);
}

__device__ __forceinline__ float gelu_exact(float x) {
    return 0.5f * x * (1.0f + erff(x * 0.70710678118654752440f));
}

// ---------------------------------------------------------------------------
// Kernel 1: embedding gather + LayerNorm stats. One wave (32 lanes) per token.
// flat[n][d] = emb[x[n]][d];  xhat = (flat - mu) * rsqrt(var + eps)
// ---------------------------------------------------------------------------
__global__ __launch_bounds__(256) void k_embed_ln(
    const int* __restrict__ x, const float* __restrict__ emb,
    float* __restrict__ flat, float* __restrict__ xhat) {
    const int wave = threadIdx.x >> 5;
    const int lane = threadIdx.x & 31;
    const int tok  = blockIdx.x * 8 + wave;            // grid = NN/8 blocks
    const int row  = x[tok];

    float4 v = ((const float4*)(emb + (size_t)row * DD))[lane];  // 4 elems/lane
    float s  = v.x + v.y + v.z + v.w;
    float sq = v.x*v.x + v.y*v.y + v.z*v.z + v.w*v.w;
    #pragma unroll
    for (int off = 16; off >= 1; off >>= 1) {
        s  += __shfl_xor(s,  off, 32);
        sq += __shfl_xor(sq, off, 32);
    }
    const float mu   = s * (1.0f / DD);
    const float var  = sq * (1.0f / DD) - mu * mu;
    const float rstd = rsqrtf(var + EPSF);

    ((float4*)(flat + (size_t)tok * DD))[lane] = v;
    float4 h;
    h.x = (v.x - mu) * rstd;  h.y = (v.y - mu) * rstd;
    h.z = (v.z - mu) * rstd;  h.w = (v.w - mu) * rstd;
    ((float4*)(xhat + (size_t)tok * DD))[lane] = h;
}

// ---------------------------------------------------------------------------
// Kernel 2: per-agent MLP for a 16-token tile, fully fused:
//   xn = xhat*g1+b1 -> GEMM1(16x128x256) -> GELU -> LN(H) -> GEMM2(16x256x128)
//   -> atomicAdd into acc (sum over agents)
// grid = (21, 256), block = 256 threads (8 waves)
// ---------------------------------------------------------------------------
__global__ __launch_bounds__(256) void k_agent_mlp(
    const float* __restrict__ xhat,
    const float* __restrict__ g1,   const float* __restrict__ b1ln,
    const float* __restrict__ W1,   const float* __restrict__ bfc1,
    const float* __restrict__ g2,   const float* __restrict__ b2ln,
    const float* __restrict__ W2,   const float* __restrict__ bfc2,
    float* __restrict__ acc) {
    __shared__ float s_xn[16 * 132];    // padded stride -> conflict-free b64 reads
    __shared__ float s_hid[16 * 260];
    __shared__ float s_sum[16][17];
    __shared__ float s_sq[16][17];
    __shared__ float s_mu[16];
    __shared__ float s_rs[16];

    const int ag      = blockIdx.x;
    const int tokbase = blockIdx.y * 16;
    const int tid     = threadIdx.x;
    const int lane    = tid & 31;
    const int wv      = tid >> 5;
    const int n       = lane & 15;      // WMMA column index for B/C/D
    const int khalf   = lane >> 4;      // 0: K/M low half, 1: high half

    // --- stage xn = xhat * g1[ag] + b1ln[ag] ---
    for (int i = tid; i < 16 * DD; i += 256) {
        const int m = i >> 7, d = i & 127;
        const float xh = xhat[(size_t)(tokbase + m) * DD + d];
        s_xn[m * 132 + d] = xh * g1[ag * DD + d] + b1ln[ag * DD + d];
    }
    __syncthreads();

    // --- GEMM1: hid = gelu(xn @ W1[ag] + bfc1[ag]); wave owns H-tiles wv, wv+8
    {
        v8f c0 = {}, c1 = {};
        const int h0a = wv * 16, h0b = (wv + 8) * 16;
        #pragma unroll 4
        for (int k4 = 0; k4 < DD / 4; ++k4) {
            const int kk = 4 * k4 + 2 * khalf;
            v2f a;
            a.x = s_xn[n * 132 + kk];
            a.y = s_xn[n * 132 + kk + 1];
            const float* bpa = W1 + ((size_t)(ag * DD + kk)) * HH + h0a + n;
            v2f b0; b0.x = bpa[0]; b0.y = bpa[HH];
            c0 = wmma_f32(a, b0, c0);
            const float* bpb = W1 + ((size_t)(ag * DD + kk)) * HH + h0b + n;
            v2f b1; b1.x = bpb[0]; b1.y = bpb[HH];
            c1 = wmma_f32(a, b1, c1);
        }
        const float biasA = bfc1[ag * HH + h0a + n];
        const float biasB = bfc1[ag * HH + h0b + n];
        const int mbase = khalf * 8;
        #pragma unroll
        for (int r = 0; r < 8; ++r) {
            const int m = mbase + r;
            s_hid[m * 260 + h0a + n] = gelu_exact(c0[r] + biasA);
            s_hid[m * 260 + h0b + n] = gelu_exact(c1[r] + biasB);
        }
    }
    __syncthreads();

    // --- LayerNorm over H=256 per token ---
    {
        const int m = tid >> 4, seg = tid & 15;
        float s = 0.f, sq = 0.f;
        #pragma unroll
        for (int j = 0; j < 16; ++j) {
            const float v = s_hid[m * 260 + seg * 16 + j];
            s += v; sq += v * v;
        }
        s_sum[m][seg] = s; s_sq[m][seg] = sq;
    }
    __syncthreads();
    if (tid < 16) {
        float s = 0.f, sq = 0.f;
        #pragma unroll
        for (int j = 0; j < 16; ++j) { s += s_sum[tid][j]; sq += s_sq[tid][j]; }
        const float mu  = s * (1.0f / HH);
        const float var = sq * (1.0f / HH) - mu * mu;
        s_mu[tid] = mu;
        s_rs[tid] = rsqrtf(var + EPSF);
    }
    __syncthreads();
    for (int i = tid; i < 16 * HH; i += 256) {
        const int m = i >> 8, h = i & 255;
        const float v = s_hid[m * 260 + h];
        s_hid[m * 260 + h] =
            (v - s_mu[m]) * s_rs[m] * g2[ag * HH + h] + b2ln[ag * HH + h];
    }
    __syncthreads();

    // --- GEMM2: out = hn @ W2[ag] + bfc2[ag]; wave owns D-tile wv ---
    {
        v8f c = {};
        const int d0 = wv * 16;
        #pragma unroll 4
        for (int k4 = 0; k4 < HH / 4; ++k4) {
            const int kk = 4 * k4 + 2 * khalf;
            v2f a;
            a.x = s_hid[n * 260 + kk];
            a.y = s_hid[n * 260 + kk + 1];
            const float* bp = W2 + ((size_t)(ag * HH + kk)) * DD + d0 + n;
            v2f b; b.x = bp[0]; b.y = bp[DD];
            c = wmma_f32(a, b, c);
        }
        const float bias = bfc2[ag * DD + d0 + n];
        const int mbase = khalf * 8;
        #pragma unroll
        for (int r = 0; r < 8; ++r) {
            const int m = mbase + r;
            atomicAdd(&acc[(size_t)(tokbase + m) * DD + d0 + n], c[r] + bias);
        }
    }
}

// ---------------------------------------------------------------------------
// Kernel 3: hidden = 2*flat + acc/21   (the per-agent +flat folds into 2*flat)
// ---------------------------------------------------------------------------
__global__ __launch_bounds__(256) void k_combine(
    const float* __restrict__ flat, const float* __restrict__ acc,
    float* __restrict__ hidden) {
    const size_t i = (size_t)blockIdx.x * 256 + threadIdx.x;
    hidden[i] = 2.0f * flat[i] + acc[i] * (1.0f / 21.0f);
}

// ---------------------------------------------------------------------------
// Kernel 4: logits = hidden[4096,128] @ Wout[128,32000] + bout
// grid = (63, 256): block covers 16 tokens x 32 vocab tiles (8 waves x 4 tiles)
// A fragment reused across 4 B tiles per K-step; logits stored nontemporal.
// ---------------------------------------------------------------------------
__global__ __launch_bounds__(256) void k_logits(
    const float* __restrict__ hidden, const float* __restrict__ Wout,
    const float* __restrict__ bout, float* __restrict__ logits) {
    __shared__ float s_h[16 * 132];

    const int tid     = threadIdx.x;
    const int lane    = tid & 31;
    const int wv      = tid >> 5;
    const int tokbase = blockIdx.y * 16;
    const int n       = lane & 15;
    const int khalf   = lane >> 4;

    for (int i = tid; i < 16 * DD; i += 256) {
        const int m = i >> 7, d = i & 127;
        s_h[m * 132 + d] = hidden[(size_t)(tokbase + m) * DD + d];
    }
    __syncthreads();

    const int ntb = blockIdx.x * 32 + wv * 4;     // first of 4 vocab tiles
    bool valid[4];
    int  col[4];
    #pragma unroll
    for (int t = 0; t < 4; ++t) {
        valid[t] = (ntb + t) < (VV / 16);         // wave-uniform predicate
        col[t]   = (ntb + t) * 16 + n;
    }

    v8f cacc[4] = {{}, {}, {}, {}};
    #pragma unroll 4
    for (int k4 = 0; k4 < DD / 4; ++k4) {
        const int kk = 4 * k4 + 2 * khalf;
        v2f a;
        a.x = s_h[n * 132 + kk];
        a.y = s_h[n * 132 + kk + 1];
        const float* brow0 = Wout + (size_t)kk * VV;
        const float* brow1 = brow0 + VV;
        #pragma unroll
        for (int t = 0; t < 4; ++t) {
            if (valid[t]) {
                v2f b; b.x = brow0[col[t]]; b.y = brow1[col[t]];
                cacc[t] = wmma_f32(a, b, cacc[t]);
            }
        }
    }

    const int mbase = khalf * 8;
    #pragma unroll
    for (int t = 0; t < 4; ++t) {
        if (!valid[t]) continue;
        const float bb = bout[col[t]];
        #pragma unroll
        for (int r = 0; r < 8; ++r) {
            const int m = mbase + r;
            __builtin_nontemporal_store(
                cacc[t][r] + bb,
                &logits[(size_t)(tokbase + m) * VV + col[t]]);
        }
    }
}

// ---------------------------------------------------------------------------
extern "C" void kernel_launch(void* const* d_in, const int* in_sizes, int n_in,
                              void* d_out, int out_size, void* d_ws, size_t ws_size,
                              hipStream_t stream) {
    const int*   x    = (const int*)  d_in[0];
    const float* emb  = (const float*)d_in[1];
    const float* Wout = (const float*)d_in[2];
    const float* bout = (const float*)d_in[3];
    const float* g1   = (const float*)d_in[4];
    const float* b1ln = (const float*)d_in[5];
    const float* W1   = (const float*)d_in[6];
    const float* bfc1 = (const float*)d_in[7];
    const float* g2   = (const float*)d_in[8];
    const float* b2ln = (const float*)d_in[9];
    const float* W2   = (const float*)d_in[10];
    const float* bfc2 = (const float*)d_in[11];

    float* ws     = (float*)d_ws;
    float* flat   = ws;                               // [N, D]
    float* xhat   = ws + 1 * (size_t)NN * DD;         // [N, D]
    float* acc    = ws + 2 * (size_t)NN * DD;         // [N, D] agent sum
    float* hidden = ws + 3 * (size_t)NN * DD;         // [N, D]
    float* logits = (float*)d_out;                    // [N, V]

    hipMemsetAsync(acc, 0, (size_t)NN * DD * sizeof(float), stream);

    k_embed_ln<<<NN / 8, 256, 0, stream>>>(x, emb, flat, xhat);

    k_agent_mlp<<<dim3(A_AG, NN / 16), 256, 0, stream>>>(
        xhat, g1, b1ln, W1, bfc1, g2, b2ln, W2, bfc2, acc);

    k_combine<<<(NN * DD) / 256, 256, 0, stream>>>(flat, acc, hidden);

    k_logits<<<dim3((VV / 16 + 31) / 32, NN / 16), 256, 0, stream>>>(
        hidden, Wout, bout, logits);
}